// trajaccumulationloss_34153579938004
// MI455X (gfx1250) — compile-verified
//
#include <hip/hip_runtime.h>
#include <hip/hip_bf16.h>
#include <stdint.h>

// ---------------------------------------------------------------------------
// Trajectory accumulation loss, MI455X (gfx1250).
// Bandwidth-bound (~75 MB -> ~3.2 us @ 23.3 TB/s). SE(2) prefix scan per row.
// CDNA5 paths: global_load_async_to_lds_b128 (ASYNCcnt) staging into the
// 320KB LDS (256KB/row), wave32 shuffle scan, LDS/global float atomics.
// WMMA is deliberately NOT used: a 16x16x4 f32 WMMA realizes only 4 useful
// diagonal 4x4 products per instruction (25% utilization) and each scan level
// would need D(16x16)->A(16x4) lane re-marshalling; the SE(2) reduction of
// the 4x4 transforms needs only ~16 VALU flops per compose and the kernel is
// memory-bound anyway.
// ---------------------------------------------------------------------------

#define L_LEN 8192
#define TPB   1024
#define EPT   (L_LEN / TPB)   // 8 contiguous elements per thread
#define WAVES (TPB / 32)      // 32 waves per block

struct SE2 { float c, s, x, z; };

__device__ __forceinline__ SE2 se2_id() { SE2 r; r.c = 1.f; r.s = 0.f; r.x = 0.f; r.z = 0.f; return r; }

// a applied first (left factor): r = A * B as matrices
__device__ __forceinline__ SE2 se2_mul(const SE2 a, const SE2 b) {
  SE2 r;
  r.c = a.c * b.c - a.s * b.s;
  r.s = a.c * b.s + a.s * b.c;
  r.x = a.x + a.c * b.x + a.s * b.z;
  r.z = a.z - a.s * b.x + a.c * b.z;
  return r;
}

__device__ __forceinline__ SE2 se2_shfl_up(const SE2 v, int d) {
  SE2 r;
  r.c = __shfl_up(v.c, d, 32);
  r.s = __shfl_up(v.s, d, 32);
  r.x = __shfl_up(v.x, d, 32);
  r.z = __shfl_up(v.z, d, 32);
  return r;
}

// wave32 inclusive Kogge-Stone scan (ordered, non-commutative op)
__device__ __forceinline__ SE2 wave_scan_incl(SE2 v, unsigned lane) {
#pragma unroll
  for (int d = 1; d < 32; d <<= 1) {
    SE2 u = se2_shfl_up(v, d);
    if ((int)lane >= d) v = se2_mul(u, v);
  }
  return v;
}

// denormalize + normalize direction -> SE(2) element
__device__ __forceinline__ SE2 make_elem(float4 raw, float4 sc, float4 mn) {
  float dx = fmaf(raw.x, sc.x, mn.x);
  float dz = fmaf(raw.y, sc.y, mn.y);
  float fx = fmaf(raw.z, sc.z, mn.z);
  float fz = fmaf(raw.w, sc.w, mn.w);
  float inv = rsqrtf(fx * fx + fz * fz);
  SE2 e; e.c = fz * inv; e.s = fx * inv; e.x = dx; e.z = dz;
  return e;
}

// CDNA5 async copy: 16B global -> LDS per lane, tracked by ASYNCcnt
__device__ __forceinline__ void async_copy_b128(unsigned lds_off, const void* gptr) {
  unsigned long long ga = (unsigned long long)(uintptr_t)gptr;
  asm volatile("global_load_async_to_lds_b128 %0, %1, off"
               :: "v"(lds_off), "v"(ga)
               : "memory");
}

__device__ __forceinline__ void wait_async0() {
  asm volatile("s_wait_asynccnt 0" ::: "memory");
}

extern __shared__ __align__(16) char smem_dyn[];

__global__ __launch_bounds__(TPB, 1)
void traj_loss_kernel(const float* __restrict__ pred,
                      const float* __restrict__ gt,
                      const int*   __restrict__ mask,
                      const float* __restrict__ mean,
                      const float* __restrict__ stdv,
                      float* __restrict__ accum /* [0]=sum, [1]=count */) {
  const int row  = blockIdx.x;
  const int tid  = threadIdx.x;
  const unsigned lane = (unsigned)(tid & 31);
  const int wave = tid >> 5;

  float4* sP = (float4*)smem_dyn;                         // L_LEN float4 (pred)
  float4* sG = (float4*)(smem_dyn + (size_t)L_LEN * 16);  // L_LEN float4 (gt)

  const float4* predRow = (const float4*)pred + (size_t)row * L_LEN;
  const float4* gtRow   = (const float4*)gt   + (size_t)row * L_LEN;
  const int4*   maskRow = (const int4*)(mask + (size_t)row * L_LEN);

  // ---- async stage: each thread copies its own contiguous chunk ----
  const int base = tid * EPT;
#pragma unroll
  for (int j = 0; j < EPT; ++j) {
    int e = base + j;
    async_copy_b128((unsigned)(uintptr_t)&sP[e], (const void*)(predRow + e));
    async_copy_b128((unsigned)(uintptr_t)&sG[e], (const void*)(gtRow + e));
  }

  // overlap with the async staging: mask (2 x b128) + denorm constants
  int4 mA = maskRow[tid * 2];
  int4 mB = maskRow[tid * 2 + 1];
  int mv[EPT] = {mA.x, mA.y, mA.z, mA.w, mB.x, mB.y, mB.z, mB.w};

  float4 mn = make_float4(mean[384], mean[385], mean[386], mean[387]);
  float4 sc = make_float4(stdv[384], stdv[385], stdv[386], stdv[387]);

  wait_async0();  // only this thread's own LDS region is read afterwards

  // ---- pass 1: thread-sequential inclusive prefixes over 8 elements ----
  // Overwrite each element's LDS slot with the thread-local inclusive prefix
  // so pass 2 needs one compose per element (no rsqrt recompute).
  SE2 aP = se2_id(), aG = se2_id();
#pragma unroll
  for (int j = 0; j < EPT; ++j) {
    int e = base + j;
    aP = se2_mul(aP, make_elem(sP[e], sc, mn));
    aG = se2_mul(aG, make_elem(sG[e], sc, mn));
    sP[e] = make_float4(aP.c, aP.s, aP.x, aP.z);
    sG[e] = make_float4(aG.c, aG.s, aG.x, aG.z);
  }

  // wave-level scan of thread aggregates
  SE2 iP = wave_scan_incl(aP, lane);
  SE2 iG = wave_scan_incl(aG, lane);
  SE2 eP = se2_shfl_up(iP, 1); if (lane == 0) eP = se2_id();
  SE2 eG = se2_shfl_up(iG, 1); if (lane == 0) eG = se2_id();

  // cross-wave scan (32 wave totals) done by wave 0
  __shared__ SE2 wTotP[WAVES], wTotG[WAVES];
  __shared__ SE2 wPreP[WAVES], wPreG[WAVES];
  if (lane == 31) { wTotP[wave] = iP; wTotG[wave] = iG; }
  __syncthreads();
  if (wave == 0) {
    SE2 tP = wTotP[lane], tG = wTotG[lane];
    SE2 sPn = wave_scan_incl(tP, lane);
    SE2 sGn = wave_scan_incl(tG, lane);
    SE2 exP = se2_shfl_up(sPn, 1); if (lane == 0) exP = se2_id();
    SE2 exG = se2_shfl_up(sGn, 1); if (lane == 0) exG = se2_id();
    wPreP[lane] = exP; wPreG[lane] = exG;
  }
  __syncthreads();

  const SE2 exAllP = se2_mul(wPreP[wave], eP);  // full exclusive prefix
  const SE2 exAllG = se2_mul(wPreG[wave], eG);

  // ---- pass 2: absolute pose per element, masked squared diff ----
  float lsum = 0.f, lcnt = 0.f;
#pragma unroll
  for (int j = 0; j < EPT; ++j) {
    int e = base + j;
    float4 p4 = sP[e];
    float4 g4 = sG[e];
    SE2 lp; lp.c = p4.x; lp.s = p4.y; lp.x = p4.z; lp.z = p4.w;
    SE2 lg; lg.c = g4.x; lg.s = g4.y; lg.x = g4.z; lg.z = g4.w;
    SE2 wp = se2_mul(exAllP, lp);
    SE2 wg = se2_mul(exAllG, lg);
    float m  = (mv[j] != 0) ? 1.f : 0.f;
    float d0 = wp.x - wg.x;   // px
    float d1 = wp.z - wg.z;   // pz
    float d2 = wp.s - wg.s;   // fx
    float d3 = wp.c - wg.c;   // fz
    lsum = fmaf(m, d0 * d0 + d1 * d1 + d2 * d2 + d3 * d3, lsum);
    lcnt += m;
  }

  // ---- hierarchical reduction ----
#pragma unroll
  for (int d = 16; d >= 1; d >>= 1) {
    lsum += __shfl_down(lsum, d, 32);
    lcnt += __shfl_down(lcnt, d, 32);
  }
  __shared__ float bSum, bCnt;
  if (tid == 0) { bSum = 0.f; bCnt = 0.f; }
  __syncthreads();
  if (lane == 0) { atomicAdd(&bSum, lsum); atomicAdd(&bCnt, lcnt); }
  __syncthreads();
  if (tid == 0) { atomicAdd(&accum[0], bSum); atomicAdd(&accum[1], bCnt); }
}

__global__ void traj_init_kernel(float* accum) {
  if (threadIdx.x < 2) accum[threadIdx.x] = 0.f;
}

__global__ void traj_finalize_kernel(const float* __restrict__ accum,
                                     float* __restrict__ out) {
  out[0] = accum[0] / (accum[1] * 4.0f);
}

extern "C" void kernel_launch(void* const* d_in, const int* in_sizes, int n_in,
                              void* d_out, int out_size, void* d_ws, size_t ws_size,
                              hipStream_t stream) {
  const float* pred = (const float*)d_in[0];
  const float* gt   = (const float*)d_in[1];
  const int*   mask = (const int*)d_in[2];
  const float* mean = (const float*)d_in[3];
  const float* stdv = (const float*)d_in[4];
  float* accum = (float*)d_ws;
  float* out   = (float*)d_out;

  const int BL = in_sizes[2];       // B * L (mask element count)
  const int B  = BL / L_LEN;

  traj_init_kernel<<<1, 32, 0, stream>>>(accum);
  const size_t shmem = (size_t)L_LEN * 16u * 2u;  // 256 KB dynamic LDS (CDNA5: 320 KB/WGP)
  traj_loss_kernel<<<B, TPB, shmem, stream>>>(pred, gt, mask, mean, stdv, accum);
  traj_finalize_kernel<<<1, 1, 0, stream>>>(accum, out);
}